// ScaledDotProductAttention_86423331930507
// MI455X (gfx1250) — compile-verified
//
#include <hip/hip_runtime.h>
#include <math.h>

// Causal scaled-dot-product attention, fp32, with full attn matrix output.
// B=16, L=4096, D=64.  d_out = [ output (B*L*D) | attn (B*L*L) ] floats.
//
// Strategy (MI455X / gfx1250):
//  * grid = (L/16, B); one 256-thread block (8 wave32) owns 16 attn rows.
//  * Pass 1: S = (Q K^T)/tau via chained v_wmma_f32_16x16x4_f32; raw scores
//    stored to the (exclusively owned) attn rows; row max via shfl + LDS.
//    Causal compares only on the diagonal tile.
//  * Pass 2: row sums of exp(s - max) -- no store (saves ~537 MB of HBM
//    writes); exp is recomputed where needed (TRANS co-executes).
//  * Pass 3: O = P V via WMMA with A-fragments exp'ed on the fly from the
//    L2-hot raw scores; final normalize writes attn with NT stores.

typedef float v2f __attribute__((ext_vector_type(2)));
typedef float v4f __attribute__((ext_vector_type(4)));
typedef float v8f __attribute__((ext_vector_type(8)));

#define ATT_L 4096
#define ATT_D 64
#define ATT_B 16
#define ATT_NEG (-1e10f)

__global__ __launch_bounds__(256) void
sdpa_causal_wmma_f32(const float* __restrict__ q,
                     const float* __restrict__ k,
                     const float* __restrict__ v,
                     const unsigned char* __restrict__ mask,
                     float* __restrict__ out,
                     float* __restrict__ attn)
{
    constexpr int L = ATT_L, D = ATT_D;
    const int qt   = blockIdx.x;          // query tile (16 rows)
    const int b    = blockIdx.y;          // batch
    const int tid  = threadIdx.x;
    const int wave = tid >> 5;            // 0..7
    const int lane = tid & 31;
    const int lm   = lane & 15;           // M (A-frag) or N (B/C-frag) index
    const int lh   = lane >> 4;           // lane half selector

    const int q0    = qt * 16;
    const int ktEnd = qt + 1;             // causal: key tiles needed
    const int Lk    = ktEnd * 16;

    const float invT = 1.0f / (8.0f + 1e-6f);

    __shared__ float         s_red[8][16];     // per-wave row maxima
    __shared__ float         s_rmax[16];
    __shared__ float         s_sum[16];
    __shared__ float         s_rinv[16];
    __shared__ unsigned char s_mask[16];
    __shared__ float         s_opart[4][256];  // key-half-1 partial O tiles

    const float* qb = q + ((size_t)b * L) * D;
    const float* kb = k + ((size_t)b * L) * D;
    const float* vb = v + ((size_t)b * L) * D;
    float* arow0 = attn + ((size_t)b * L + q0) * L;  // 16 rows, stride L
    float* ob    = out  + ((size_t)b * L + q0) * D;

    if (tid < 16) {
        s_mask[tid] = mask[(size_t)b * L + q0 + tid];
        s_sum[tid]  = 0.0f;
    }
    // zero the strictly-above-causal region attn[:, Lk:L]  (never re-read: NT)
    {
        const int z4 = (L - Lk) >> 2;
        const v4f zv = {0.f, 0.f, 0.f, 0.f};
        for (int m = 0; m < 16; ++m) {
            v4f* zp = (v4f*)(arow0 + (size_t)m * L + Lk);
            for (int i = tid; i < z4; i += 256)
                __builtin_nontemporal_store(zv, zp + i);
        }
    }
    __syncthreads();

    // ---------------- Pass 1: S = (Q K^T)/tau, mask, store raw, row max
    // Q A-fragments: A 16x4 f32 -> lanes 0-15 hold K=0,1; lanes 16-31 K=2,3
    v2f qfrag[16];
    {
        const float* qr = qb + (size_t)(q0 + lm) * D + 2 * lh;
#pragma unroll
        for (int t = 0; t < 16; ++t)
            qfrag[t] = *(const v2f*)(qr + 4 * t);
    }
    unsigned pmbits = 0;
#pragma unroll
    for (int j = 0; j < 8; ++j)
        if (s_mask[j + 8 * lh]) pmbits |= (1u << j);

    float wmax[8];
#pragma unroll
    for (int j = 0; j < 8; ++j) wmax[j] = -3.0e38f;

    // full (strictly below-diagonal) tiles: pad mask only, no causal compare
    for (int kt = wave; kt < qt; kt += 8) {
        const int k0 = kt * 16;
        const float* kr = kb + (size_t)(k0 + lm) * D + 2 * lh;  // B-frag of K^T
        v8f c = {};
#pragma unroll
        for (int t = 0; t < 16; ++t) {
            v2f kf = *(const v2f*)(kr + 4 * t);
            c = __builtin_amdgcn_wmma_f32_16x16x4_f32(
                    false, qfrag[t], false, kf, (short)0, c, false, false);
        }
        const int kcol = k0 + lm;
#pragma unroll
        for (int j = 0; j < 8; ++j) {
            float s = c[j] * invT;
            s = ((pmbits >> j) & 1u) ? ATT_NEG : s;
            arow0[(size_t)(j + 8 * lh) * L + kcol] = s;
            wmax[j] = fmaxf(wmax[j], s);
        }
    }
    // diagonal tile: one wave, with causal compare
    if (wave == (qt & 7)) {
        const int k0 = qt * 16;
        const float* kr = kb + (size_t)(k0 + lm) * D + 2 * lh;
        v8f c = {};
#pragma unroll
        for (int t = 0; t < 16; ++t) {
            v2f kf = *(const v2f*)(kr + 4 * t);
            c = __builtin_amdgcn_wmma_f32_16x16x4_f32(
                    false, qfrag[t], false, kf, (short)0, c, false, false);
        }
        const int kcol = k0 + lm;
#pragma unroll
        for (int j = 0; j < 8; ++j) {
            const int qrow = q0 + j + 8 * lh;
            float s = c[j] * invT;
            const bool bad = (kcol > qrow) || ((pmbits >> j) & 1u);
            s = bad ? ATT_NEG : s;
            arow0[(size_t)(j + 8 * lh) * L + kcol] = s;
            wmax[j] = fmaxf(wmax[j], s);
        }
    }
    // reduce row maxima across the 16 lanes of each half, then across waves
#pragma unroll
    for (int j = 0; j < 8; ++j) {
        float m = wmax[j];
        m = fmaxf(m, __shfl_xor(m, 1, 32));
        m = fmaxf(m, __shfl_xor(m, 2, 32));
        m = fmaxf(m, __shfl_xor(m, 4, 32));
        m = fmaxf(m, __shfl_xor(m, 8, 32));
        wmax[j] = m;
    }
    if (lm == 0) {
#pragma unroll
        for (int j = 0; j < 8; ++j)
            s_red[wave][j + 8 * lh] = wmax[j];
    }
    __syncthreads();
    if (tid < 16) {
        float m = s_red[0][tid];
#pragma unroll
        for (int w = 1; w < 8; ++w) m = fmaxf(m, s_red[w][tid]);
        s_rmax[tid] = m;
    }
    __syncthreads();

    // ---------------- Pass 2: row sums of exp(s - rowmax); NO store
    for (int m = 0; m < 16; ++m) {
        const float rmax = s_rmax[m];
        const v4f* prow = (const v4f*)(arow0 + (size_t)m * L);
        const int n4 = Lk >> 2;
        float part = 0.0f;
        for (int i = tid; i < n4; i += 256) {
            v4f sv = prow[i];
            part += (__expf(sv.x - rmax) + __expf(sv.y - rmax)) +
                    (__expf(sv.z - rmax) + __expf(sv.w - rmax));
        }
        part += __shfl_xor(part, 1, 32);
        part += __shfl_xor(part, 2, 32);
        part += __shfl_xor(part, 4, 32);
        part += __shfl_xor(part, 8, 32);
        part += __shfl_xor(part, 16, 32);
        if (lane == 0) atomicAdd(&s_sum[m], part);
    }
    __syncthreads();
    if (tid < 16) s_rinv[tid] = 1.0f / s_sum[tid];
    __syncthreads();

    // ---------------- Pass 3: O = (P V) * rinv, P = exp(raw - rmax) on the fly
    const int dg   = wave & 3;        // 16-wide D column group
    const int half = wave >> 2;       // key half
    const int d0   = dg * 16;
    const int hT   = (ktEnd + 1) >> 1;
    const int t0   = half ? hT : 0;
    const int t1   = half ? ktEnd : hT;
    const float rmaxA = s_rmax[lm];   // A-frag rows: M = lane%16

    v8f acc = {};
    for (int kt = t0; kt < t1; ++kt) {
#pragma unroll
        for (int c4 = 0; c4 < 4; ++c4) {
            const int kk = kt * 16 + c4 * 4;
            // A-frag: raw scores -> p, 2 TRANS ops per WMMA (co-executed)
            v2f sraw = *(const v2f*)(arow0 + (size_t)lm * L + kk + 2 * lh);
            v2f a;
            a.x = __expf(sraw.x - rmaxA);
            a.y = __expf(sraw.y - rmaxA);
            // B-frag: V[4 x 16], column d0+lm fixed per lane
            v2f bf;
            bf.x = vb[(size_t)(kk + 2 * lh) * D + d0 + lm];
            bf.y = vb[(size_t)(kk + 2 * lh + 1) * D + d0 + lm];
            acc = __builtin_amdgcn_wmma_f32_16x16x4_f32(
                      false, a, false, bf, (short)0, acc, false, false);
        }
    }
    if (wave >= 4) {
#pragma unroll
        for (int j = 0; j < 8; ++j)
            s_opart[dg][j * 32 + lane] = acc[j];
    }
    __syncthreads();  // orders all raw-score reads before the rewrite below
    if (wave < 4) {
#pragma unroll
        for (int j = 0; j < 8; ++j) {
            const int m = j + 8 * lh;
            float o = (acc[j] + s_opart[dg][j * 32 + lane]) * s_rinv[m];
            __builtin_nontemporal_store(o, ob + (size_t)m * D + d0 + lm);
        }
    }
    // final normalize: last read of raw scores (NT), NT store of attn
    for (int m = 0; m < 16; ++m) {
        const float rmax = s_rmax[m];
        const float ri   = s_rinv[m];
        v4f* prow = (v4f*)(arow0 + (size_t)m * L);
        const int n4 = Lk >> 2;
        for (int i = tid; i < n4; i += 256) {
            v4f sv = __builtin_nontemporal_load(prow + i);
            v4f pv;
            pv.x = __expf(sv.x - rmax) * ri;
            pv.y = __expf(sv.y - rmax) * ri;
            pv.z = __expf(sv.z - rmax) * ri;
            pv.w = __expf(sv.w - rmax) * ri;
            __builtin_nontemporal_store(pv, prow + i);
        }
    }
}

extern "C" void kernel_launch(void* const* d_in, const int* in_sizes, int n_in,
                              void* d_out, int out_size, void* d_ws, size_t ws_size,
                              hipStream_t stream) {
    const float*         q    = (const float*)d_in[0];
    const float*         k    = (const float*)d_in[1];
    const float*         v    = (const float*)d_in[2];
    const unsigned char* mask = (const unsigned char*)d_in[3];

    float* out  = (float*)d_out;                                  // [B,L,D]
    float* attn = out + (size_t)ATT_B * ATT_L * ATT_D;            // [B,L,L]

    dim3 grid(ATT_L / 16, ATT_B);
    sdpa_causal_wmma_f32<<<grid, 256, 0, stream>>>(q, k, v, mask, out, attn);
}